// CU_50259707298079
// MI455X (gfx1250) — compile-verified
//
#include <hip/hip_runtime.h>
#include <math.h>

typedef __attribute__((ext_vector_type(2))) float v2f;
typedef __attribute__((ext_vector_type(4))) float v4f;
typedef __attribute__((ext_vector_type(8))) float v8f;

// ---------------------------------------------------------------------------
// Kernel 1: closed-form expm of the 2x2 anti-Hermitian generator A = M - M^H.
// A = i*H (H Hermitian) => exp(A) = e^{i*s} (cos(r) I + sinc(r) * B),
// where s = Im(tr(M)), B = A - i*s*I (traceless), B^2 = -r^2 I.
// Writes Mexp as 8 floats {e00r,e00i,e01r,e01i,e10r,e10i,e11r,e11i} to ws.
// ---------------------------------------------------------------------------
__global__ void expm2x2_kernel(const float* __restrict__ M_re,
                               const float* __restrict__ M_im,
                               float* __restrict__ mex) {
    if (threadIdx.x != 0 || blockIdx.x != 0) return;
    float m00i = M_im[0], m01r = M_re[1], m01i = M_im[1];
    float m10r = M_re[2], m10i = M_im[2], m11i = M_im[3];
    // A = M - M^H: A00 = 2i*Im(M00), A11 = 2i*Im(M11), A01 = M01 - conj(M10)
    float s_im = m00i + m11i;          // tr(A)/2 = i*s_im
    float bb   = m00i - m11i;          // B00 = i*bb, B11 = -i*bb
    float a01r = m01r - m10r;          // B01 = A01
    float a01i = m01i + m10i;
    float r   = sqrtf(bb*bb + a01r*a01r + a01i*a01i);
    float cr  = cosf(r);
    float snc = (r > 1e-20f) ? (sinf(r) / r) : 1.0f;
    float cp  = cosf(s_im), sp = sinf(s_im);   // e^{i*s_im}
    float sb  = snc * bb;
    // Mexp00 = e^{is}(cr + i*sb), Mexp11 = e^{is}(cr - i*sb)
    mex[0] = cp*cr - sp*sb;   mex[1] = sp*cr + cp*sb;   // e00
    mex[6] = cp*cr + sp*sb;   mex[7] = sp*cr - cp*sb;   // e11
    // Mexp01 = e^{is} * snc * A01 ; Mexp10 = e^{is} * snc * (-conj(A01))
    mex[2] = snc*(cp*a01r - sp*a01i);  mex[3] = snc*(sp*a01r + cp*a01i); // e01
    float b10r = -a01r, b10i = a01i;
    mex[4] = snc*(cp*b10r - sp*b10i);  mex[5] = snc*(sp*b10r + cp*b10i); // e10
}

// ---------------------------------------------------------------------------
// Kernel 2: identity half (rows 0..2047): interleave re/im, b128 stores.
// Each thread handles 2 complex elements: 2x b64 loads, 1x b128 store.
// ---------------------------------------------------------------------------
__global__ void copy_top_kernel(const float* __restrict__ xr,
                                const float* __restrict__ xi,
                                float* __restrict__ out) {
    long gid  = (long)blockIdx.x * blockDim.x + threadIdx.x; // 1,048,576 threads
    long base = gid * 2;                                     // element index
    v2f re = *(const v2f*)(xr + base);
    v2f im = *(const v2f*)(xi + base);
    v4f o  = { re.x, im.x, re.y, im.y };
    *(v4f*)(out + base * 2) = o;
}

// ---------------------------------------------------------------------------
// Kernel 3: controlled-Mexp half (rows 2048..4095) via V_WMMA_F32_16X16X4_F32.
//   out = R(4x4 real) @ X(4 x 1M), tiled 16 columns per WMMA.
// A (16x4): rows 0..3 = R, rest 0.  Lane layout: lane=M, K0/K1 in VGPR0/1 for
// lanes<16, K2/K3 for lanes>=16.
// B (4x16): lane<16 holds (x_top_re, x_top_im) = K0/K1 for column N=lane;
//           lane>=16 holds (x_bot_re, x_bot_im) = K2/K3 for column N=lane-16.
// D: VGPR0..3 lanes 0..15 = rows M=0..3 = (ot_re, ot_im, ob_re, ob_im).
// Software-pipelined: 4 tiles/wave -> 8 loads in one clause, 4 independent
// WMMAs back-to-back, one predicated store clause (512B contiguous/half-wave).
// ---------------------------------------------------------------------------
__global__ void cu_bottom_wmma_kernel(const float* __restrict__ xr,
                                      const float* __restrict__ xi,
                                      const float* __restrict__ mex,
                                      float* __restrict__ out) {
    const int lane = threadIdx.x & 31;
    const int wave = blockIdx.x * (blockDim.x >> 5) + (threadIdx.x >> 5);

    // Build A operand once per wave (uniform per lane).
    // R[m][k]: p=m>>1 (out block), q=k>>1 (in block), m&1 / k&1 = re/im parts:
    //   R = [[er,-ei],[ei,er]] per 2x2 complex entry e_{pq}.
    const int m  = lane & 15;
    const int kb = (lane < 16) ? 0 : 2;
    v2f a = { 0.0f, 0.0f };
    if (m < 4) {
        const int p = m >> 1, mc = m & 1;
        #pragma unroll
        for (int t = 0; t < 2; ++t) {
            const int k = kb + t, q = k >> 1, kc = k & 1;
            const float er = mex[(p * 2 + q) * 2 + 0];
            const float ei = mex[(p * 2 + q) * 2 + 1];
            const float v = (mc == 0) ? ((kc == 0) ? er : -ei)
                                      : ((kc == 0) ? ei :  er);
            if (t == 0) a.x = v; else a.y = v;
        }
    }

    // Tiles wave*4 .. wave*4+3 share the same row-pair j (wave*4 % 64 <= 60),
    // and advance by 16 columns each -> constant base + 64B immediate offsets.
    const int tile0 = wave * 4;                // 65536 tiles total
    const int j     = tile0 >> 6;              // 0..1023
    const int c0    = (tile0 & 63) << 4;       // column tile base
    const int col   = c0 + (lane & 15);
    const int row   = (lane < 16) ? (2048 + j) : (3072 + j);
    const int idx   = row * 1024 + col;

    // Stage 1: issue all 8 loads (one clause), fill 4 B operands.
    v2f b[4];
    #pragma unroll
    for (int t = 0; t < 4; ++t) {
        b[t].x = xr[idx + 16 * t];
        b[t].y = xi[idx + 16 * t];
    }

    // Stage 2: 4 independent WMMAs (distinct accumulators, no RAW hazard).
    v8f acc[4];
    #pragma unroll
    for (int t = 0; t < 4; ++t) {
        v8f z = { 0.f, 0.f, 0.f, 0.f, 0.f, 0.f, 0.f, 0.f };
        acc[t] = __builtin_amdgcn_wmma_f32_16x16x4_f32(
                     /*neg_a=*/false, a, /*neg_b=*/false, b[t],
                     /*c_mod=*/(short)0, z, /*reuse_a=*/false, /*reuse_b=*/false);
    }

    // Stage 3: single predicated region, one store clause.
    if (lane < 16) {
        const long ot = ((long)(2048 + j) * 1024 + col) * 2;
        const long ob = ((long)(3072 + j) * 1024 + col) * 2;
        #pragma unroll
        for (int t = 0; t < 4; ++t) {
            v2f top = { acc[t][0], acc[t][1] };   // M=0,1: out_top re/im
            v2f bot = { acc[t][2], acc[t][3] };   // M=2,3: out_bot re/im
            *(v2f*)(out + ot + 32 * t) = top;     // +16 cols = +32 floats
            *(v2f*)(out + ob + 32 * t) = bot;
        }
    }
}

extern "C" void kernel_launch(void* const* d_in, const int* in_sizes, int n_in,
                              void* d_out, int out_size, void* d_ws, size_t ws_size,
                              hipStream_t stream) {
    const float* M_re = (const float*)d_in[0];   // 2x2
    const float* M_im = (const float*)d_in[1];   // 2x2
    const float* x_re = (const float*)d_in[2];   // 4096x1024
    const float* x_im = (const float*)d_in[3];   // 4096x1024
    float* out = (float*)d_out;                  // 4096x1024 complex64 -> interleaved f32
    float* mex = (float*)d_ws;                   // 8 floats: Mexp

    (void)in_sizes; (void)n_in; (void)out_size; (void)ws_size;

    // 1) Mexp = expm(M - M^H), closed form (tiny).
    expm2x2_kernel<<<1, 32, 0, stream>>>(M_re, M_im, mex);

    // 2) Top half: identity copy w/ re-im interleave. 2048*1024 elems,
    //    2 per thread -> 1,048,576 threads.
    copy_top_kernel<<<4096, 256, 0, stream>>>(x_re, x_im, out);

    // 3) Bottom half: 2x2 complex block-apply via WMMA f32 16x16x4.
    //    65536 tiles / (8 waves * 4 tiles) = 2048 blocks.
    cu_bottom_wmma_kernel<<<2048, 256, 0, stream>>>(x_re, x_im, mex, out);
}